// basic_block_69054484185132
// MI455X (gfx1250) — compile-verified
//
#include <hip/hip_runtime.h>
#include <math.h>

#define SLOPE  0.01f
#define BN_EPS 1e-5f

#define B_SZ   16
#define N_SZ   128
#define DIN    128
#define DOUT   64
#define ODIM   256
#define MROWS  (B_SZ * N_SZ)           /* 2048 */
#define NPAIR  (B_SZ * N_SZ * N_SZ)    /* 262144 */

typedef float v2f __attribute__((ext_vector_type(2)));
typedef float v8f __attribute__((ext_vector_type(8)));

__device__ __forceinline__ v8f wmma_f32_16x16x4(v2f a, v2f b, v8f c) {
  return __builtin_amdgcn_wmma_f32_16x16x4_f32(false, a, false, b, (short)0, c, false, false);
}

__device__ __forceinline__ float lrelu(float z) { return z >= 0.f ? z : SLOPE * z; }

__device__ __forceinline__ int adj_ch(int l) {
  // ADJ_CH = [256, 192, 192, 96, 96, 1]
  return (l == 0) ? 256 : (l == 1) ? 192 : (l == 2) ? 192
       : (l == 3) ? 96  : (l == 4) ? 96  : 1;
}

// ---------------------------------------------------------------------------
// Batched GEMM: C[b] = A[b] (Mdim x K, row-major) @ Bm[b] (K x N, row-major)
// One wave per 16x32 output tile (dual column tiles share the A fragment).
// ---------------------------------------------------------------------------
__global__ __launch_bounds__(32)
void bmm_kernel(const float* __restrict__ A, const float* __restrict__ Bm,
                float* __restrict__ C, int Mdim, int K, int N) {
  const int tn = N >> 5;                 // 32-wide column tiles
  const int tm = Mdim >> 4;
  int t = blockIdx.x;
  const int n0 = (t % tn) << 5; t /= tn;
  const int m0 = (t % tm) << 4;
  const int b  = t / tm;
  const int lane = threadIdx.x;
  const int m = lane & 15, half = lane >> 4, koff = half << 1;
  const float* Ab = A  + (size_t)b * Mdim * K;
  const float* Bb = Bm + (size_t)b * K * N;
  float*       Cb = C  + (size_t)b * Mdim * N;
  v8f acc0 = {}, acc1 = {};
  for (int k0 = 0; k0 < K; k0 += 4) {
    v2f av = *(const v2f*)(Ab + (size_t)(m0 + m) * K + k0 + koff);
    const float* brow0 = Bb + (size_t)(k0 + koff) * N + n0 + m;
    const float* brow1 = Bb + (size_t)(k0 + koff + 1) * N + n0 + m;
    v2f bv0, bv1;
    bv0.x = brow0[0];  bv0.y = brow1[0];
    bv1.x = brow0[16]; bv1.y = brow1[16];
    acc0 = wmma_f32_16x16x4(av, bv0, acc0);
    acc1 = wmma_f32_16x16x4(av, bv1, acc1);
  }
#pragma unroll
  for (int r = 0; r < 8; ++r) {
    Cb[(size_t)(m0 + r + half * 8) * N + n0 + m]      = acc0[r];
    Cb[(size_t)(m0 + r + half * 8) * N + n0 + 16 + m] = acc1[r];
  }
}

// ---------------------------------------------------------------------------
// Linear: Z (M x Cout) = X (M x K) @ W^T (W is [Cout][K] row-major) + bias
// One wave per 16x32 output tile.
// ---------------------------------------------------------------------------
__global__ __launch_bounds__(32)
void linear_kernel(const float* __restrict__ X, const float* __restrict__ W,
                   const float* __restrict__ bias, float* __restrict__ Z,
                   int M, int K, int Cout) {
  const int tn = Cout >> 5;
  const int n0   = (blockIdx.x % tn) << 5;
  const int row0 = (blockIdx.x / tn) << 4;
  const int lane = threadIdx.x;
  const int m = lane & 15, half = lane >> 4, koff = half << 1;
  (void)M;
  v8f acc0 = {}, acc1 = {};
  for (int k0 = 0; k0 < K; k0 += 4) {
    v2f av  = *(const v2f*)(X + (size_t)(row0 + m) * K + k0 + koff);
    v2f bv0 = *(const v2f*)(W + (size_t)(n0 + m)      * K + k0 + koff);
    v2f bv1 = *(const v2f*)(W + (size_t)(n0 + 16 + m) * K + k0 + koff);
    acc0 = wmma_f32_16x16x4(av, bv0, acc0);
    acc1 = wmma_f32_16x16x4(av, bv1, acc1);
  }
  const float bb0 = bias[n0 + m];
  const float bb1 = bias[n0 + 16 + m];
#pragma unroll
  for (int r = 0; r < 8; ++r) {
    Z[(size_t)(row0 + r + half * 8) * Cout + n0 + m]      = acc0[r] + bb0;
    Z[(size_t)(row0 + r + half * 8) * Cout + n0 + 16 + m] = acc1[r] + bb1;
  }
}

// ---------------------------------------------------------------------------
// Small BN stats (over M rows, channel-last [M][C]) -> scale/shift per channel
// ---------------------------------------------------------------------------
__global__ void stats_small_kernel(const float* __restrict__ Z, int M, int C,
                                   const float* __restrict__ gamma,
                                   const float* __restrict__ beta,
                                   float* __restrict__ scale,
                                   float* __restrict__ shift) {
  const int c = blockIdx.x;
  __shared__ float ssum[256];
  __shared__ float ssq[256];
  float s = 0.f, s2 = 0.f;
  for (int r = threadIdx.x; r < M; r += blockDim.x) {
    float z = Z[(size_t)r * C + c];
    s += z; s2 += z * z;
  }
  ssum[threadIdx.x] = s; ssq[threadIdx.x] = s2;
  __syncthreads();
  for (int off = 128; off > 0; off >>= 1) {
    if ((int)threadIdx.x < off) {
      ssum[threadIdx.x] += ssum[threadIdx.x + off];
      ssq[threadIdx.x]  += ssq[threadIdx.x + off];
    }
    __syncthreads();
  }
  if (threadIdx.x == 0) {
    float mean = ssum[0] / (float)M;
    float var  = ssq[0] / (float)M - mean * mean;
    float sc   = gamma[c] * rsqrtf(var + BN_EPS);
    scale[c] = sc;
    shift[c] = beta[c] - mean * sc;
  }
}

__global__ void apply_bn_lrelu_kernel(const float* __restrict__ Z,
                                      const float* __restrict__ scale,
                                      const float* __restrict__ shift,
                                      float* __restrict__ out,
                                      int M, int C, int ldout, int colOff) {
  int idx = blockIdx.x * blockDim.x + threadIdx.x;
  if (idx >= M * C) return;
  int r = idx / C, c = idx % C;
  out[(size_t)r * ldout + colOff + c] = lrelu(Z[idx] * scale[c] + shift[c]);
}

__global__ void copy_cols_kernel(const float* __restrict__ X, float* __restrict__ out,
                                 int M, int C, int ldout) {
  int idx = blockIdx.x * blockDim.x + threadIdx.x;
  if (idx >= M * C) return;
  int r = idx / C, c = idx % C;
  out[(size_t)r * ldout + c] = X[idx];
}

__global__ void xout_kernel(const float* __restrict__ xcat, const float* __restrict__ ztr,
                            const float* __restrict__ scale, const float* __restrict__ shift,
                            float* __restrict__ out, int total, int C) {
  int idx = blockIdx.x * blockDim.x + threadIdx.x;
  if (idx >= total) return;
  int c = idx % C;
  out[idx] = lrelu(xcat[idx] + ztr[idx] * scale[c] + shift[c]);
}

// ---------------------------------------------------------------------------
// BN finalize for the edge-MLP layers + fold scale into weights/bias.
// ---------------------------------------------------------------------------
__global__ void finalize_fold_kernel(const double* __restrict__ stats,
                                     const float* __restrict__ W,
                                     const float* __restrict__ bvec,
                                     const float* __restrict__ gamma,
                                     const float* __restrict__ beta,
                                     float* __restrict__ Wf, float* __restrict__ bf,
                                     int Cin, float invCnt) {
  const int c = blockIdx.x;
  const float mean = (float)(stats[c * 2 + 0] * (double)invCnt);
  const float ex2  = (float)(stats[c * 2 + 1] * (double)invCnt);
  const float var  = ex2 - mean * mean;
  const float sc   = gamma[c] * rsqrtf(var + BN_EPS);
  const float sh   = beta[c] - mean * sc;
  for (int k = threadIdx.x; k < Cin; k += blockDim.x)
    Wf[(size_t)c * Cin + k] = W[(size_t)c * Cin + k] * sc;
  if (threadIdx.x == 0) bf[c] = bvec[c] * sc + sh;
}

// ---------------------------------------------------------------------------
// Fused edge-MLP pass. One wave per (b, i, 16-j slab).
// stat_layer 0..3: folded layers < stat_layer, f64 stats of raw layer.
// stat_layer 4: full folded chain + conv_last + sigmoid + diag mask -> Aout.
// All layer widths are multiples of 32 -> dual column tiles per A fragment.
// ---------------------------------------------------------------------------
__global__ __launch_bounds__(32)
void acompute_pass_kernel(const float* __restrict__ xcat,
                          const float* __restrict__ Wf,
                          const float* __restrict__ bf,
                          const float* __restrict__ Wraw,
                          const float* __restrict__ braw,
                          double* __restrict__ stats,
                          float* __restrict__ Aout,
                          int stat_layer) {
  __shared__ __align__(16) float bufA[16][260];
  __shared__ __align__(16) float bufB[16][200];

  const int wg   = blockIdx.x;
  const int j0   = (wg & 7) << 4;
  const int i    = (wg >> 3) & (N_SZ - 1);
  const int b    = wg >> 10;
  const int lane = threadIdx.x;
  const int m    = lane & 15;
  const int half = lane >> 4;
  const int koff = half << 1;

  // features |x_i - x_j| for 16 consecutive j
  const float* xb = xcat + (size_t)b * N_SZ * ODIM;
  const float* xi = xb + (size_t)i * ODIM;
  for (int idx = lane; idx < 16 * ODIM; idx += 32) {
    int r = idx >> 8;
    int c = idx & (ODIM - 1);
    bufA[r][c] = fabsf(xi[c] - xb[(size_t)(j0 + r) * ODIM + c]);
  }
  __syncthreads();

  float* bin  = &bufA[0][0]; int ldin  = 260;
  float* bout = &bufB[0][0]; int ldout = 200;

  const float* Wl = Wf;
  const float* bl = bf;
  int cin = ODIM;
  const int nfold = (stat_layer < 4) ? stat_layer : 4;
  for (int l = 0; l < nfold; ++l) {
    const int cout = adj_ch(l + 1);
    for (int n0 = 0; n0 < cout; n0 += 32) {
      v8f acc0 = {}, acc1 = {};
      for (int k0 = 0; k0 < cin; k0 += 4) {
        v2f av  = *(const v2f*)(bin + m * ldin + k0 + koff);
        v2f bv0 = *(const v2f*)(Wl + (size_t)(n0 + m)      * cin + k0 + koff);
        v2f bv1 = *(const v2f*)(Wl + (size_t)(n0 + 16 + m) * cin + k0 + koff);
        acc0 = wmma_f32_16x16x4(av, bv0, acc0);
        acc1 = wmma_f32_16x16x4(av, bv1, acc1);
      }
      const float bias0 = bl[n0 + m];
      const float bias1 = bl[n0 + 16 + m];
#pragma unroll
      for (int r = 0; r < 8; ++r) {
        bout[(r + half * 8) * ldout + n0 + m]      = lrelu(acc0[r] + bias0);
        bout[(r + half * 8) * ldout + n0 + 16 + m] = lrelu(acc1[r] + bias1);
      }
    }
    __syncthreads();
    float* tp = bin; bin = bout; bout = tp;
    int   tl = ldin; ldin = ldout; ldout = tl;
    Wl += (size_t)cin * cout;
    bl += cout;
    cin = cout;
  }

  if (stat_layer < 4) {
    const int cout = adj_ch(stat_layer + 1);
    for (int n0 = 0; n0 < cout; n0 += 32) {
      v8f acc0 = {}, acc1 = {};
      for (int k0 = 0; k0 < cin; k0 += 4) {
        v2f av  = *(const v2f*)(bin + m * ldin + k0 + koff);
        v2f bv0 = *(const v2f*)(Wraw + (size_t)(n0 + m)      * cin + k0 + koff);
        v2f bv1 = *(const v2f*)(Wraw + (size_t)(n0 + 16 + m) * cin + k0 + koff);
        acc0 = wmma_f32_16x16x4(av, bv0, acc0);
        acc1 = wmma_f32_16x16x4(av, bv1, acc1);
      }
      const float bias0 = braw[n0 + m];
      const float bias1 = braw[n0 + 16 + m];
      double s0 = 0.0, s0q = 0.0, s1 = 0.0, s1q = 0.0;
#pragma unroll
      for (int r = 0; r < 8; ++r) {
        double z0 = (double)(acc0[r] + bias0);
        double z1 = (double)(acc1[r] + bias1);
        s0 += z0; s0q += z0 * z0;
        s1 += z1; s1q += z1 * z1;
      }
      s0  += __shfl_xor(s0, 16, 32);
      s0q += __shfl_xor(s0q, 16, 32);
      s1  += __shfl_xor(s1, 16, 32);
      s1q += __shfl_xor(s1q, 16, 32);
      if (half == 0) {
        atomicAdd(&stats[(n0 + m) * 2 + 0], s0);
        atomicAdd(&stats[(n0 + m) * 2 + 1], s0q);
        atomicAdd(&stats[(n0 + 16 + m) * 2 + 0], s1);
        atomicAdd(&stats[(n0 + 16 + m) * 2 + 1], s1q);
      }
    }
  } else {
    // conv_last: 96 -> 1, then sigmoid, then zero the diagonal
    float s = 0.f;
    const int kbase = half * 48;
    for (int k = 0; k < 48; ++k) s += bin[m * ldin + kbase + k] * Wraw[kbase + k];
    s += __shfl_xor(s, 16, 32);
    if (half == 0) {
      float z = s + braw[0];
      float a = 1.f / (1.f + __expf(-z));
      if (i == j0 + m) a = 0.f;
      Aout[((size_t)(b * N_SZ + i)) * N_SZ + j0 + m] = a;
    }
  }
}

// ---------------------------------------------------------------------------
// D[b,i] = sqrt(1 + sum_j A[b,i,j])   (A has zero diagonal)
// ---------------------------------------------------------------------------
__global__ void rowsum_kernel(const float* __restrict__ A, float* __restrict__ D) {
  const int row = blockIdx.x;           // b*N + i
  __shared__ float sh[128];
  sh[threadIdx.x] = A[(size_t)row * N_SZ + threadIdx.x];
  __syncthreads();
  for (int off = 64; off > 0; off >>= 1) {
    if ((int)threadIdx.x < off) sh[threadIdx.x] += sh[threadIdx.x + off];
    __syncthreads();
  }
  if (threadIdx.x == 0) D[row] = sqrtf(1.0f + sh[0]);
}

__global__ void normalize_kernel(const float* __restrict__ A, const float* __restrict__ D,
                                 float* __restrict__ out) {
  int idx = blockIdx.x * blockDim.x + threadIdx.x;
  if (idx >= NPAIR) return;
  int j  = idx & (N_SZ - 1);
  int bi = idx >> 7;
  int b  = bi >> 7;
  int i  = bi & (N_SZ - 1);
  float a = A[idx] + ((i == j) ? 1.f : 0.f);
  out[idx] = a / (D[bi] * D[b * N_SZ + j]);
}

// ---------------------------------------------------------------------------
// Host launch
// ---------------------------------------------------------------------------
extern "C" void kernel_launch(void* const* d_in, const int* in_sizes, int n_in,
                              void* d_out, int out_size, void* d_ws, size_t ws_size,
                              hipStream_t stream) {
  (void)in_sizes; (void)n_in; (void)out_size; (void)ws_size;

  // Inputs in setup_inputs() insertion order (nested dicts flattened in order)
  const float* x      = (const float*)d_in[0];
  const float* Anorm  = (const float*)d_in[1];
  const float* W_b1g0 = (const float*)d_in[2];
  const float* b_b1g0 = (const float*)d_in[3];
  const float* g_b1g0 = (const float*)d_in[4];
  const float* e_b1g0 = (const float*)d_in[5];
  const float* W_b2g0 = (const float*)d_in[6];
  const float* b_b2g0 = (const float*)d_in[7];
  const float* g_b2g0 = (const float*)d_in[8];
  const float* e_b2g0 = (const float*)d_in[9];
  const float* W_b2g1 = (const float*)d_in[10];
  const float* b_b2g1 = (const float*)d_in[11];
  const float* g_b2g1 = (const float*)d_in[12];
  const float* e_b2g1 = (const float*)d_in[13];
  const float* Wa[5]  = {(const float*)d_in[14], (const float*)d_in[18],
                         (const float*)d_in[22], (const float*)d_in[26],
                         (const float*)d_in[30]};
  const float* ba[5]  = {(const float*)d_in[15], (const float*)d_in[19],
                         (const float*)d_in[23], (const float*)d_in[27],
                         (const float*)d_in[31]};
  const float* ga[4]  = {(const float*)d_in[16], (const float*)d_in[20],
                         (const float*)d_in[24], (const float*)d_in[28]};
  const float* ea[4]  = {(const float*)d_in[17], (const float*)d_in[21],
                         (const float*)d_in[25], (const float*)d_in[29]};
  const float* W_tr   = (const float*)d_in[32];
  const float* b_tr   = (const float*)d_in[33];
  const float* g_bn   = (const float*)d_in[34];
  const float* e_bn   = (const float*)d_in[35];

  // --- workspace layout ---
  const size_t STATS_BYTES = 4 * 384 * sizeof(double);      // 4 layers x 192ch x {sum,sumsq}
  double* statsD = (double*)d_ws;
  float*  fb     = (float*)((char*)d_ws + STATS_BYTES);
  float* T0   = fb;                         // 262144
  float* z1   = T0  + 262144;               // 131072
  float* z2a  = z1  + 131072;               // 131072
  float* h2a  = z2a + 131072;               // 131072
  float* T1   = h2a + 131072;               // 131072
  float* z2   = T1  + 131072;               // 131072
  float* xcat = z2  + 131072;               // 524288
  float* ztr  = xcat + 524288;              // 524288
  float* Wf   = ztr + 524288;               // 113664 (folded edge-MLP weights)
  float* bfv  = Wf  + 113664;               // 576
  float* sc1  = bfv + 576;    float* sh1  = sc1 + 64;
  float* sc2a = sh1 + 64;     float* sh2a = sc2a + 64;
  float* sc2  = sh2a + 64;    float* sh2  = sc2 + 64;
  float* scT  = sh2 + 64;     float* shT  = scT + 256;
  float* Dvec = shT + 256;                  // 2048

  const size_t WFOFF[4] = {0, 49152, 86016, 104448};
  const size_t BFOFF[4] = {0, 192, 384, 480};

  float* A_out  = (float*)d_out;            // 262144
  float* An_out = A_out + NPAIR;            // 262144
  float* X_out  = An_out + NPAIR;           // 524288

  hipMemsetAsync((void*)statsD, 0, STATS_BYTES, stream);

  // --- branch gconvs ---
  bmm_kernel<<<B_SZ * 8 * 4, 32, 0, stream>>>(Anorm, x, T0, N_SZ, DIN, DIN);       // T0 = A@x
  linear_kernel<<<(MROWS / 16) * (DOUT / 32), 32, 0, stream>>>(T0, W_b1g0, b_b1g0, z1, MROWS, DIN, DOUT);
  linear_kernel<<<(MROWS / 16) * (DOUT / 32), 32, 0, stream>>>(T0, W_b2g0, b_b2g0, z2a, MROWS, DIN, DOUT);
  stats_small_kernel<<<DOUT, 256, 0, stream>>>(z1,  MROWS, DOUT, g_b1g0, e_b1g0, sc1,  sh1);
  stats_small_kernel<<<DOUT, 256, 0, stream>>>(z2a, MROWS, DOUT, g_b2g0, e_b2g0, sc2a, sh2a);
  apply_bn_lrelu_kernel<<<(MROWS * DOUT + 255) / 256, 256, 0, stream>>>(z1,  sc1,  sh1,  xcat, MROWS, DOUT, ODIM, 128);
  apply_bn_lrelu_kernel<<<(MROWS * DOUT + 255) / 256, 256, 0, stream>>>(z2a, sc2a, sh2a, h2a,  MROWS, DOUT, DOUT, 0);
  bmm_kernel<<<B_SZ * 8 * 2, 32, 0, stream>>>(Anorm, h2a, T1, N_SZ, DIN, DOUT);    // T1 = A@h2a
  linear_kernel<<<(MROWS / 16) * (DOUT / 32), 32, 0, stream>>>(T1, W_b2g1, b_b2g1, z2, MROWS, DOUT, DOUT);
  stats_small_kernel<<<DOUT, 256, 0, stream>>>(z2, MROWS, DOUT, g_b2g1, e_b2g1, sc2, sh2);
  apply_bn_lrelu_kernel<<<(MROWS * DOUT + 255) / 256, 256, 0, stream>>>(z2, sc2, sh2, xcat, MROWS, DOUT, ODIM, 192);
  copy_cols_kernel<<<(MROWS * DIN + 255) / 256, 256, 0, stream>>>(x, xcat, MROWS, DIN, ODIM);

  // --- trans branch + x_out ---
  linear_kernel<<<(MROWS / 16) * (ODIM / 32), 32, 0, stream>>>(x, W_tr, b_tr, ztr, MROWS, DIN, ODIM);
  stats_small_kernel<<<ODIM, 256, 0, stream>>>(ztr, MROWS, ODIM, g_bn, e_bn, scT, shT);
  xout_kernel<<<(MROWS * ODIM + 255) / 256, 256, 0, stream>>>(xcat, ztr, scT, shT, X_out, MROWS * ODIM, ODIM);

  // --- fused edge-MLP: 4 stats passes + 1 output pass ---
  const int NSLAB = B_SZ * N_SZ * (N_SZ / 16);   // 16384
  const float invCnt = 1.0f / (float)NPAIR;
  for (int l = 0; l < 4; ++l) {
    acompute_pass_kernel<<<NSLAB, 32, 0, stream>>>(xcat, Wf, bfv, Wa[l], ba[l],
                                                   statsD + l * 384, A_out, l);
    const int cout = (l < 2) ? 192 : 96;
    const int cin  = (l == 0) ? 256 : (l == 1) ? 192 : (l == 2) ? 192 : 96;
    finalize_fold_kernel<<<cout, 128, 0, stream>>>(statsD + l * 384, Wa[l], ba[l],
                                                   ga[l], ea[l],
                                                   Wf + WFOFF[l], bfv + BFOFF[l],
                                                   cin, invCnt);
  }
  acompute_pass_kernel<<<NSLAB, 32, 0, stream>>>(xcat, Wf, bfv, Wa[4], ba[4],
                                                 statsD, A_out, 4);

  // --- normalization outputs ---
  rowsum_kernel<<<MROWS, 128, 0, stream>>>(A_out, Dvec);
  normalize_kernel<<<(NPAIR + 255) / 256, 256, 0, stream>>>(A_out, Dvec, An_out);
}